// DeformConv_29231547417116
// MI455X (gfx1250) — compile-verified
//
#include <hip/hip_runtime.h>
#include <math.h>

typedef __attribute__((ext_vector_type(2))) float v2f;
typedef __attribute__((ext_vector_type(4))) float v4f;
typedef __attribute__((ext_vector_type(8))) float v8f;

#define Bn   4
#define Cc   128
#define Hh   128
#define Ww   128
#define CO   128
#define HW   (Hh*Ww)
#define KDIM (Cc*9)      // 1152
#define ASTRIDE 148      // LDS row stride in floats: stride%64==20 -> conflict-free b64 reads

// ---------------------------------------------------------------------------
// Pass 0: pack weights into WMMA B-fragment order so each wave issues one
// coalesced b128 load per lane covering TWO 16x16x4 steps.
//   layout: [n_tile][t = K/8][lane][4]
//     j in 0..3: step s = 2t + (j>>1); K = 4s + 2*(lane>>4) + (j&1); N = n_tile*16 + lane%16
// ---------------------------------------------------------------------------
__global__ __launch_bounds__(256) void pack_weights(
    const float* __restrict__ w, const float* __restrict__ w_om,
    float* __restrict__ Wtb, float* __restrict__ Wtomb)
{
    int id = blockIdx.x * 256 + threadIdx.x;
    if (id < 8 * 144 * 128) {                    // main conv: 8 n-tiles
        int j = id & 3, lane = (id >> 2) & 31;
        int t = (id >> 7) % 144, ng = (id >> 7) / 144;
        int s = 2 * t + (j >> 1);
        int K = 4 * s + ((lane >> 4) << 1) + (j & 1);
        int N = ng * 16 + (lane & 15);
        Wtb[id] = w[N * KDIM + K];
    } else {
        int id2 = id - 8 * 144 * 128;            // offset conv: 2 n-tiles (27 pad 32)
        if (id2 < 2 * 144 * 128) {
            int j = id2 & 3, lane = (id2 >> 2) & 31;
            int t = (id2 >> 7) % 144, ng = (id2 >> 7) / 144;
            int s = 2 * t + (j >> 1);
            int K = 4 * s + ((lane >> 4) << 1) + (j & 1);
            int N = ng * 16 + (lane & 15);
            Wtomb[id2] = (N < 27) ? w_om[N * KDIM + K] : 0.0f;
        }
    }
}

// ---------------------------------------------------------------------------
// Pass 1: offset/mask conv (im2col GEMM, N=27 pad 32) + sample transform.
// Grid: B*H*2. Tile M=64 along w, 8 waves = (mi 0..3) x (ni 0..1).
// Double-buffered A in LDS, one barrier per K-chunk.
// ---------------------------------------------------------------------------
__global__ __launch_bounds__(256) void offset_conv(
    const float* __restrict__ x, const float* __restrict__ Wtomb,
    const float* __restrict__ b_om,
    float* __restrict__ pyA, float* __restrict__ pxA, float* __restrict__ mkA)
{
    __shared__ __align__(16) float As[2][64 * ASTRIDE];   // 75.8 KB
    const int wg   = blockIdx.x;
    const int tile = wg & 1;
    const int h    = (wg >> 1) & (Hh - 1);
    const int b    = wg >> 8;
    const int w0   = tile * 64;

    const int tid  = threadIdx.x;
    const int lane = tid & 31;
    const int wave = tid >> 5;
    const int mi   = wave >> 1;
    const int ni   = wave & 1;
    const int nlo  = lane & 15;
    const int khalf = (lane >> 4) * 2;

    auto buildA = [&](int ci, float* dst) {
        for (int i = 0; i < 36; ++i) {
            int e  = i * 256 + tid;
            int m  = e & 63;
            int kk = e >> 6;                     // 0..143
            int c  = ci * 16 + kk / 9;
            int k2 = kk % 9;
            int yy = h + (k2 / 3) - 1;
            int xx = w0 + m + (k2 % 3) - 1;
            float v = 0.0f;
            if ((unsigned)yy < (unsigned)Hh && (unsigned)xx < (unsigned)Ww)
                v = x[(size_t)(b * Cc + c) * HW + yy * Ww + xx];
            dst[m * ASTRIDE + kk] = v;
        }
    };

    v8f acc = {};
    buildA(0, As[0]);
    __syncthreads();
    for (int ci = 0; ci < 8; ++ci) {
        if (ci < 7) buildA(ci + 1, As[(ci + 1) & 1]);
        const float* Bp = Wtomb + (size_t)((ni * 144 + ci * 18) * 128) + lane * 4;
        const float* Ap = &As[ci & 1][(mi * 16 + nlo) * ASTRIDE + khalf];
#pragma unroll
        for (int tt = 0; tt < 18; ++tt) {
            v4f bq = *(const v4f*)(Bp + tt * 128);
            v2f a0 = *(const v2f*)(Ap + 8 * tt);
            v2f a1 = *(const v2f*)(Ap + 8 * tt + 4);
            v2f b0 = __builtin_shufflevector(bq, bq, 0, 1);
            v2f b1 = __builtin_shufflevector(bq, bq, 2, 3);
            acc = __builtin_amdgcn_wmma_f32_16x16x4_f32(
                false, a0, false, b0, (short)0, acc, false, false);
            acc = __builtin_amdgcn_wmma_f32_16x16x4_f32(
                false, a1, false, b1, (short)0, acc, false, false);
        }
        __syncthreads();
    }

    // epilogue: + b_om, round-trip through LDS, compute py/px/mask
    int ch = ni * 16 + nlo;
    float bias = (ch < 27) ? b_om[ch] : 0.0f;
    float* om = As[0];                           // reuse: 64 x 33 floats
#pragma unroll
    for (int r = 0; r < 8; ++r) {
        int m = mi * 16 + r + ((lane >> 4) << 3);
        om[m * 33 + ch] = acc[r] + bias;
    }
    __syncthreads();
    for (int t = tid; t < 576; t += 256) {
        int m  = t & 63;
        int k2 = t >> 6;                         // 0..8
        float dy = om[m * 33 + 2 * k2];
        float dx = om[m * 33 + 2 * k2 + 1];
        float ms = 1.0f / (1.0f + expf(-om[m * 33 + 18 + k2]));
        int gw = w0 + m;
        size_t s = (size_t)((b * Hh + h) * Ww + gw) * 9 + k2;
        pyA[s] = (float)(h - 1 + k2 / 3) + dy;
        pxA[s] = (float)(gw - 1 + k2 % 3) + dx;
        mkA[s] = ms;
    }
}

// ---------------------------------------------------------------------------
// Pass 2: deformable GEMM (M=16 spatial x N=128) + bias + BN + ReLU.
// Grid: B*H*(W/16) = 4096 blocks, 8 waves each own one N-tile of 16.
// Double-buffered A in LDS; bilinear gathers overlap the WMMA stream.
// ---------------------------------------------------------------------------
__global__ __launch_bounds__(256) void deform_gemm(
    const float* __restrict__ x,  const float* __restrict__ Wtb,
    const float* __restrict__ bc, const float* __restrict__ gamma,
    const float* __restrict__ beta, const float* __restrict__ bnm,
    const float* __restrict__ bnv,
    const float* __restrict__ pyA, const float* __restrict__ pxA,
    const float* __restrict__ mkA, float* __restrict__ out)
{
    __shared__ int   midx[4][144];
    __shared__ float mwgt[4][144];
    __shared__ __align__(16) float As[2][16 * ASTRIDE];   // 18.9 KB

    const int wg   = blockIdx.x;
    const int tile = wg & 7;
    const int h    = (wg >> 3) & (Hh - 1);
    const int b    = wg >> 10;
    const int w0   = tile * 16;

    const int tid  = threadIdx.x;
    const int lane = tid & 31;
    const int wave = tid >> 5;
    const int nlo  = lane & 15;
    const int khalf = (lane >> 4) * 2;
    const int n0   = wave * 16;

    // per-tile sample metadata: 16 positions x 9 taps -> 4 corners (idx, wgt*mask)
    if (tid < 144) {
        int m = tid / 9, k2 = tid % 9;
        size_t s = (size_t)((b * Hh + h) * Ww + (w0 + m)) * 9 + k2;
        float py = pyA[s], px = pxA[s], ms = mkA[s];
        float y0f = floorf(py), x0f = floorf(px);
        float fy = py - y0f, fx = px - x0f;
        int y0 = (int)y0f, x0 = (int)x0f;
#pragma unroll
        for (int cy = 0; cy < 2; ++cy)
#pragma unroll
            for (int cx = 0; cx < 2; ++cx) {
                int yy = y0 + cy, xx = x0 + cx;
                bool v = (unsigned)yy < (unsigned)Hh && (unsigned)xx < (unsigned)Ww;
                midx[cy * 2 + cx][tid] = v ? (yy * Ww + xx) : 0;
                mwgt[cy * 2 + cx][tid] =
                    v ? ms * (cy ? fy : 1.0f - fy) * (cx ? fx : 1.0f - fx) : 0.0f;
            }
    }
    __syncthreads();

    auto buildA = [&](int ci, float* dst) {
        for (int i = 0; i < 9; ++i) {
            int e  = i * 256 + tid;
            int m  = e & 15;
            int kk = e >> 4;                     // 0..143
            int c  = ci * 16 + kk / 9;
            int k2 = kk % 9;
            int mk = m * 9 + k2;
            const float* plane = x + (size_t)(b * Cc + c) * HW;
            dst[m * ASTRIDE + kk] =
                  mwgt[0][mk] * plane[midx[0][mk]]
                + mwgt[1][mk] * plane[midx[1][mk]]
                + mwgt[2][mk] * plane[midx[2][mk]]
                + mwgt[3][mk] * plane[midx[3][mk]];
        }
    };

    v8f acc = {};
    buildA(0, As[0]);
    __syncthreads();
    for (int ci = 0; ci < 8; ++ci) {
        if (ci < 7) buildA(ci + 1, As[(ci + 1) & 1]);
        const float* Bp = Wtb + (size_t)((wave * 144 + ci * 18) * 128) + lane * 4;
        const float* Ap = &As[ci & 1][nlo * ASTRIDE + khalf];
#pragma unroll
        for (int tt = 0; tt < 18; ++tt) {
            v4f bq = *(const v4f*)(Bp + tt * 128);
            v2f a0 = *(const v2f*)(Ap + 8 * tt);
            v2f a1 = *(const v2f*)(Ap + 8 * tt + 4);
            v2f b0 = __builtin_shufflevector(bq, bq, 0, 1);
            v2f b1 = __builtin_shufflevector(bq, bq, 2, 3);
            acc = __builtin_amdgcn_wmma_f32_16x16x4_f32(
                false, a0, false, b0, (short)0, acc, false, false);
            acc = __builtin_amdgcn_wmma_f32_16x16x4_f32(
                false, a1, false, b1, (short)0, acc, false, false);
        }
        __syncthreads();
    }

    // epilogue: y = (acc + bias - mean)*gamma/sqrt(var+eps) + beta, ReLU
    int o = n0 + nlo;
    float sc = gamma[o] * rsqrtf(bnv[o] + 1e-5f);
    float sh = beta[o] + (bc[o] - bnm[o]) * sc;
    float* ot   = &As[0][0];                     // 2*16*148 = 4736 floats available
    float* myot = ot + wave * 320;               // 16 rows x stride 20 (16B aligned)
#pragma unroll
    for (int r = 0; r < 8; ++r) {
        float v = acc[r] * sc + sh;
        v = v > 0.0f ? v : 0.0f;
        int m = r + ((lane >> 4) << 3);
        myot[nlo * 20 + m] = v;                  // [o_sub][m]
    }
    __syncthreads();
#pragma unroll
    for (int it = 0; it < 2; ++it) {
        int q = it * 32 + lane;                  // 0..63 quads of the 16x16 tile
        int osub = q >> 2, seg = q & 3;
        v4f v = *(const v4f*)&myot[osub * 20 + seg * 4];
        *(v4f*)&out[((size_t)(b * CO + n0 + osub) * Hh + h) * Ww + w0 + seg * 4] = v;
    }
}

// ---------------------------------------------------------------------------
extern "C" void kernel_launch(void* const* d_in, const int* in_sizes, int n_in,
                              void* d_out, int out_size, void* d_ws, size_t ws_size,
                              hipStream_t stream)
{
    const float* x      = (const float*)d_in[0];
    const float* w_om   = (const float*)d_in[1];
    const float* b_om   = (const float*)d_in[2];
    const float* w      = (const float*)d_in[3];
    const float* bconv  = (const float*)d_in[4];
    const float* gamma  = (const float*)d_in[5];
    const float* beta   = (const float*)d_in[6];
    const float* bn_mean= (const float*)d_in[7];
    const float* bn_var = (const float*)d_in[8];
    float* out = (float*)d_out;

    // workspace (floats): Wtb 147456 | Wtomb 36864 | py | px | mk (each 65536*9)
    float* Wtb   = (float*)d_ws;
    float* Wtomb = Wtb   + 8 * 144 * 128;
    float* pyA   = Wtomb + 2 * 144 * 128;
    float* pxA   = pyA  + (size_t)Bn * Hh * Ww * 9;
    float* mkA   = pxA  + (size_t)Bn * Hh * Ww * 9;

    int nT = (8 * 144 * 128 + 2 * 144 * 128) / 256;     // 720 blocks exactly
    pack_weights<<<nT, 256, 0, stream>>>(w, w_om, Wtb, Wtomb);
    offset_conv<<<Bn * Hh * 2, 256, 0, stream>>>(x, Wtomb, b_om, pyA, pxA, mkA);
    deform_gemm<<<Bn * Hh * (Ww / 16), 256, 0, stream>>>(
        x, Wtb, bconv, gamma, beta, bn_mean, bn_var, pyA, pxA, mkA, out);
}